// MaxPool2d_11012296147200
// MI455X (gfx1250) — compile-verified
//
#include <hip/hip_runtime.h>
#include <stdint.h>
#include <math.h>

// MaxPool2d 3x3/stride2 VALID, NCHW: X(32,64,224,224) f32 -> Y(32,64,111,111) f32.
// Streaming / HBM-bound (~512 MB => ~22us at 23.3 TB/s). Strategy: per block,
// TDM-load a 9x224 fp32 tile into LDS (input footprint of 4 output rows),
// wait on TENSORcnt, then compute 4x111 outputs from LDS (stride-2 LDS reads are
// bank-conflict-free in wave32 with 64 banks).

#define H  224
#define W  224
#define OH 111
#define OW 111
#define ROWS_PER_TILE 4                 // output rows per block
#define IN_ROWS 9                       // 2*ROWS_PER_TILE + 1 input rows
#define NTILES ((OH + ROWS_PER_TILE - 1) / ROWS_PER_TILE)   // 28
#define NCH 64
#define NB  32

typedef uint32_t v4u __attribute__((ext_vector_type(4)));
typedef int32_t  v8i __attribute__((ext_vector_type(8)));
typedef int32_t  v4i __attribute__((ext_vector_type(4)));

__global__ __launch_bounds__(128) void maxpool3x3s2_kernel(const float* __restrict__ X,
                                                           float* __restrict__ Y) {
    __shared__ float tile[IN_ROWS * W];     // 9*224*4 = 8064 B of the 320 KB/WGP LDS

    const int tid = threadIdx.x;
    const int t   = blockIdx.x;             // output-row tile
    const int c   = blockIdx.y;
    const int b   = blockIdx.z;
    const int r0      = t * ROWS_PER_TILE;  // first output row of this tile
    const int in_row0 = r0 * 2;             // first input row needed

    const uint64_t plane = (uint64_t)(b * NCH + c);
    const uint64_t gbyte = (uint64_t)(uintptr_t)X +
                           ((plane * H + (uint64_t)in_row0) * (uint64_t)W) * 4ull;
    const uint32_t remaining = (uint32_t)(H - in_row0);          // rows left in plane
    const uint32_t trows     = remaining < IN_ROWS ? remaining : IN_ROWS;

#if __has_builtin(__builtin_amdgcn_tensor_load_to_lds)
    if (tid == 0) {
        // ---- Tensor DMA descriptor (CDNA5 ISA) ----
        // Group 0: [1:0]=count=1 | [63:32]=lds_addr | [120:64]=global_addr | [127:126]=type=2
        v4u g0;
        g0.x = 1u;                                        // count=1, user mode
        g0.y = (uint32_t)(uintptr_t)(&tile[0]);           // LDS byte address
        g0.z = (uint32_t)gbyte;                           // global_addr[31:0]
        g0.w = (uint32_t)(gbyte >> 32) | (2u << 30);      // global_addr[56:32] | type=2
        // Group 1: mask=0 | data_size=2(4B) | dims/tile/stride
        v8i g1;
        g1[0] = (int)(2u << 16);                          // data_size = 4 bytes
        g1[1] = (int)((uint32_t)W << 16);                 // tensor_dim0 low16 = 224
        g1[2] = (int)(remaining << 16);                   // dim0 hi=0 | tensor_dim1 low16
        g1[3] = (int)((uint32_t)W << 16);                 // dim1 hi=0 | tile_dim0 = 224
        g1[4] = (int)trows;                               // tile_dim1 (<=9) | tile_dim2=0
        g1[5] = W;                                        // tensor_dim0_stride low32 = 224
        g1[6] = 0;                                        // stride0 hi | stride1 low
        g1[7] = 0;                                        // stride1 hi
        v4i gz4 = {0, 0, 0, 0};                           // groups 2/3 unused (2D tensor)
        v8i gz8 = {0, 0, 0, 0, 0, 0, 0, 0};               // trailing group (clang-23 form)
        __builtin_amdgcn_tensor_load_to_lds(g0, g1, gz4, gz4, gz8, 0);
        __builtin_amdgcn_s_wait_tensorcnt(0);             // wave 0 waits for the DMA
    }
#else
    // Fallback: cooperative coalesced staging into LDS.
    {
        const float* src = X + (size_t)plane * H * W + (size_t)in_row0 * W;
        const int total = IN_ROWS * W;
        const int avail = (int)remaining * W;
        for (int i = tid; i < total; i += (int)blockDim.x)
            tile[i] = (i < avail) ? src[i] : -INFINITY;
    }
#endif
    __syncthreads();                                      // release all 4 waves

    if (tid < OW) {
        const int q2 = 2 * tid;                           // stride-2: conflict-free on 64 banks
#pragma unroll
        for (int i = 0; i < ROWS_PER_TILE; ++i) {
            const int r = r0 + i;
            if (r < OH) {
                const float* row0 = &tile[(2 * i + 0) * W + q2];
                const float* row1 = &tile[(2 * i + 1) * W + q2];
                const float* row2 = &tile[(2 * i + 2) * W + q2];
                float m = row0[0];
                m = fmaxf(m, row0[1]); m = fmaxf(m, row0[2]);
                m = fmaxf(m, row1[0]); m = fmaxf(m, row1[1]); m = fmaxf(m, row1[2]);
                m = fmaxf(m, row2[0]); m = fmaxf(m, row2[1]); m = fmaxf(m, row2[2]);
                Y[(plane * OH + (uint64_t)r) * OW + (uint64_t)tid] = m;
            }
        }
    }
}

extern "C" void kernel_launch(void* const* d_in, const int* in_sizes, int n_in,
                              void* d_out, int out_size, void* d_ws, size_t ws_size,
                              hipStream_t stream) {
    const float* X = (const float*)d_in[0];
    float*       Y = (float*)d_out;
    (void)in_sizes; (void)n_in; (void)out_size; (void)d_ws; (void)ws_size;
    dim3 grid(NTILES, NCH, NB);   // 28 x 64 x 32 = 57,344 blocks
    maxpool3x3s2_kernel<<<grid, 128, 0, stream>>>(X, Y);
}